// GNNStack_87935160418397
// MI455X (gfx1250) — compile-verified
//
#include <hip/hip_runtime.h>

// ---------------------------------------------------------------------------
// GCN stack for MI455X (gfx1250, wave32). All dense transforms run on the
// matrix cores via V_WMMA_F32_16X16X4_F32 (exact fp32 semantics). The edge
// aggregation is L2-resident (working set << 192MB) scatter-add with f32
// atomics. 50000 % 16 == 0, so GEMM tiles are always full (EXEC all-ones as
// WMMA requires).
// ---------------------------------------------------------------------------

typedef __attribute__((ext_vector_type(2))) float v2f;
typedef __attribute__((ext_vector_type(8))) float v8f;

#define NNODES 50000
#define DH     128
#define DOUT   16
#define NEDGES 800000

// ---------------- degree / normalization ----------------------------------
__global__ void deg_init_kernel(float* deg, int n) {
    int i = blockIdx.x * blockDim.x + threadIdx.x;
    if (i < n) deg[i] = 1.0f;                 // self-loop contributes 1
}

__global__ void deg_count_kernel(const int* __restrict__ dst, float* deg, int e) {
    int i = blockIdx.x * blockDim.x + threadIdx.x;
    if (i < e) atomicAdd(&deg[dst[i]], 1.0f);
}

__global__ void deg_rsqrt_kernel(float* deg, int n) {
    int i = blockIdx.x * blockDim.x + threadIdx.x;
    if (i < n) deg[i] = rsqrtf(fmaxf(deg[i], 1.0f));   // deg -> d^{-1/2}
}

// ---------------- WMMA GEMM: OUT[M,N] = X[M,128] @ W[128,N] (+bias) --------
// NT = number of 16-column tiles (N = 16*NT). One wave computes a 16x(16*NT)
// strip. A-fragment (16x4 f32): lane L holds row L%16, K = k + 2*(L/16) .. +1.
// B-fragment (4x16 f32): lane L holds col L%16, same K pair. W is staged in
// LDS with (k,k+1) pairs interleaved per column so B loads are ds_load_b64.
template <int NT>
__global__ __launch_bounds__(256) void gemm_wmma_kernel(
    const float* __restrict__ X, const float* __restrict__ W,
    const float* __restrict__ bias, float* __restrict__ OUT, int M)
{
    constexpr int K = 128;
    constexpr int N = NT * 16;
    __shared__ float Wl[K * N];   // paired layout: [(k>>1)][col][k&1]

    for (int idx = threadIdx.x; idx < K * N; idx += 256) {
        int k = idx / N, col = idx - k * N;
        Wl[(k >> 1) * (2 * N) + col * 2 + (k & 1)] = W[idx];
    }
    __syncthreads();

    const int wave    = threadIdx.x >> 5;
    const int lane    = threadIdx.x & 31;
    const int rowTile = blockIdx.x * 8 + wave;
    if (rowTile * 16 >= M) return;            // uniform per-wave exit
    const int row0 = rowTile * 16;
    const int r    = lane & 15;               // row (A) / col (B,C)
    const int kh   = lane >> 4;               // which K pair of the 4

    const float* xrow = X + (size_t)(row0 + r) * K;

    v8f acc[NT];
    const v8f vzero = {0.f, 0.f, 0.f, 0.f, 0.f, 0.f, 0.f, 0.f};
#pragma unroll
    for (int t = 0; t < NT; ++t) acc[t] = vzero;

    for (int k = 0; k < K; k += 4) {
        const int kb = k + 2 * kh;
        v2f a = *(const v2f*)(xrow + kb);     // contiguous 8B load
#pragma unroll
        for (int t = 0; t < NT; ++t) {
            v2f b = *(const v2f*)(&Wl[(kb >> 1) * (2 * N) + (t * 16 + r) * 2]);
            acc[t] = __builtin_amdgcn_wmma_f32_16x16x4_f32(
                false, a, false, b, (short)0, acc[t], false, false);
        }
    }

    // C/D layout: VGPR j -> row row0 + j + 8*kh, col = t*16 + r
#pragma unroll
    for (int t = 0; t < NT; ++t) {
        const int col = t * 16 + r;
        const float bv = bias ? bias[col] : 0.0f;
#pragma unroll
        for (int j = 0; j < 8; ++j) {
            const int rr = row0 + j + 8 * kh;
            OUT[(size_t)rr * N + col] = acc[t][j] + bv;
        }
    }
}

// ---------------- self-loop + bias init ------------------------------------
// out[n,c] = b[c] + h[n,c] * dinv[n]^2   (self-loop term of A_hat)
__global__ void selfloop_bias_kernel(const float* __restrict__ h,
                                     const float* __restrict__ dinv,
                                     const float* __restrict__ bias,
                                     float* __restrict__ out, int n, int c)
{
    int i = blockIdx.x * blockDim.x + threadIdx.x;
    if (i >= n * c) return;
    int node = i / c;
    int col  = i - node * c;
    float dv = dinv[node];
    out[i] = bias[col] + h[i] * dv * dv;
}

// ---------------- edge scatter-add -----------------------------------------
// One wave per edge: 32 lanes x float4 = 128 features. All traffic L2-hot.
__global__ __launch_bounds__(256) void scatter_kernel(
    const float* __restrict__ h, const float* __restrict__ dinv,
    const int* __restrict__ src, const int* __restrict__ dst,
    float* __restrict__ out, int e)
{
    int edge = blockIdx.x * 8 + (threadIdx.x >> 5);
    if (edge >= e) return;
    int lane = threadIdx.x & 31;
    int s = src[edge], d = dst[edge];
    float nrm = dinv[s] * dinv[d];
    float4 v = ((const float4*)(h + (size_t)s * DH))[lane];
    float* op = out + (size_t)d * DH + lane * 4;
    atomicAdd(op + 0, v.x * nrm);
    atomicAdd(op + 1, v.y * nrm);
    atomicAdd(op + 2, v.z * nrm);
    atomicAdd(op + 3, v.w * nrm);
}

// ---------------- log-softmax over 16 classes ------------------------------
__global__ void log_softmax16_kernel(const float* __restrict__ logits,
                                     float* __restrict__ out, int n)
{
    int i = blockIdx.x * blockDim.x + threadIdx.x;
    if (i >= n) return;
    const float* rw = logits + (size_t)i * DOUT;
    float m = rw[0];
#pragma unroll
    for (int c = 1; c < DOUT; ++c) m = fmaxf(m, rw[c]);
    float s = 0.0f;
#pragma unroll
    for (int c = 0; c < DOUT; ++c) s += expf(rw[c] - m);
    float ls = m + logf(s);
    float* ow = out + (size_t)i * DOUT;
#pragma unroll
    for (int c = 0; c < DOUT; ++c) ow[c] = rw[c] - ls;
}

// ---------------------------------------------------------------------------
extern "C" void kernel_launch(void* const* d_in, const int* in_sizes, int n_in,
                              void* d_out, int out_size, void* d_ws, size_t ws_size,
                              hipStream_t stream)
{
    (void)in_sizes; (void)n_in; (void)out_size; (void)ws_size;

    const float* x   = (const float*)d_in[0];
    const int*   ei  = (const int*)d_in[1];   // [2, E] int32 (harness canonicalizes int)
    const float* W1  = (const float*)d_in[2];
    const float* b1  = (const float*)d_in[3];
    const float* W2  = (const float*)d_in[4];
    const float* b2  = (const float*)d_in[5];
    const float* W3  = (const float*)d_in[6];
    const float* b3  = (const float*)d_in[7];
    const float* Wp1 = (const float*)d_in[8];
    const float* bp1 = (const float*)d_in[9];
    const float* Wp2 = (const float*)d_in[10];
    const float* bp2 = (const float*)d_in[11];

    const int* src = ei;             // row 0
    const int* dst = ei + NEDGES;    // row 1

    float* ws     = (float*)d_ws;
    float* dinv   = ws;                                   // 50000 (pad to 51200)
    float* h      = ws + 51200;                           // 6.4M
    float* bufA   = h + (size_t)NNODES * DH;              // 6.4M
    float* logits = bufA + (size_t)NNODES * DH;           // 0.8M

    float* emb = (float*)d_out;                           // [N,128]
    float* lsm = (float*)d_out + (size_t)NNODES * DH;     // [N,16]

    const dim3 blk(256);
    const int gN   = (NNODES + 255) / 256;
    const int gE   = (NEDGES + 255) / 256;
    const int gNC  = (NNODES * DH + 255) / 256;
    const int gSc  = (NEDGES + 7) / 8;                    // 8 edges (waves) / block
    const int gGem = (NNODES / 16 + 7) / 8;               // 3125 tiles / 8 waves

    // degrees -> d^{-1/2}
    deg_init_kernel<<<gN, blk, 0, stream>>>(dinv, NNODES);
    deg_count_kernel<<<gE, blk, 0, stream>>>(dst, dinv, NEDGES);
    deg_rsqrt_kernel<<<gN, blk, 0, stream>>>(dinv, NNODES);

    // layer 1
    gemm_wmma_kernel<8><<<gGem, blk, 0, stream>>>(x, W1, nullptr, h, NNODES);
    selfloop_bias_kernel<<<gNC, blk, 0, stream>>>(h, dinv, b1, bufA, NNODES, DH);
    scatter_kernel<<<gSc, blk, 0, stream>>>(h, dinv, src, dst, bufA, NEDGES);

    // layer 2
    gemm_wmma_kernel<8><<<gGem, blk, 0, stream>>>(bufA, W2, nullptr, h, NNODES);
    selfloop_bias_kernel<<<gNC, blk, 0, stream>>>(h, dinv, b2, bufA, NNODES, DH);
    scatter_kernel<<<gSc, blk, 0, stream>>>(h, dinv, src, dst, bufA, NEDGES);

    // layer 3 -> embedding directly into d_out
    gemm_wmma_kernel<8><<<gGem, blk, 0, stream>>>(bufA, W3, nullptr, h, NNODES);
    selfloop_bias_kernel<<<gNC, blk, 0, stream>>>(h, dinv, b3, emb, NNODES, DH);
    scatter_kernel<<<gSc, blk, 0, stream>>>(h, dinv, src, dst, emb, NEDGES);

    // MLP head: (emb @ Wp1 + bp1) @ Wp2 + bp2, then log_softmax
    gemm_wmma_kernel<8><<<gGem, blk, 0, stream>>>(emb, Wp1, bp1, h, NNODES);
    gemm_wmma_kernel<1><<<gGem, blk, 0, stream>>>(h, Wp2, bp2, logits, NNODES);
    log_softmax16_kernel<<<gN, blk, 0, stream>>>(logits, lsm, NNODES);
}